// DigitCaps_85847806313123
// MI455X (gfx1250) — compile-verified
//
#include <hip/hip_runtime.h>
#include <math.h>

// DigitCaps dynamic routing, fused single kernel for gfx1250 (MI455X).
// Shapes: x[B=256, N=1152, I=8], W[C=10, N, I, O=16] -> out[B, C, O] (f32).
//
// Algebraic reduction: only the LAST iteration's v is returned, and logit
// updates need only r[b,n] = sum_i x[b,n,i] * Wr[c,n,i], Wr = sum_o W.
// So we do: r -> softmax -> b2 -> softmax -> ONE weighted vote GEMM via
// V_WMMA_F32_16X16X4_F32 (routing weight folded into A operand per lane),
// then squash. The 189 MB vote tensor u is never materialized.

typedef float v2f __attribute__((ext_vector_type(2)));
typedef float v8f __attribute__((ext_vector_type(8)));

#define BB 256
#define NN 1152
#define II 8
#define OO 16
#define CC 10
#define BT 16          // b-rows per workgroup (WMMA M dimension)
#define THREADS 256    // 8 waves of 32
#define NPW (NN / 8)   // 144 route nodes per wave in the WMMA pass

__global__ __launch_bounds__(THREADS)
void digitcaps_route_wmma(const float* __restrict__ x,
                          const float* __restrict__ W,
                          float* __restrict__ out)
{
    __shared__ float sWr[NN * II];      // 36864 B: Wr[n*8+i] = sum_o W[c,n,i,o]
    __shared__ float sRedM[16][16];     // per-b softmax max reduction / reused as s[M][o]
    __shared__ float sRedZ[16][16];     // per-b softmax sum reduction
    __shared__ float sM1[16], sZ1[16], sM2[16], sZ2[16];
    __shared__ float sAcc[8][8][32];    // 8192 B: per-wave WMMA accumulators

    const int   tid   = threadIdx.x;
    const int   c     = blockIdx.x >> 4;          // 0..9
    const int   bbase = (blockIdx.x & 15) * BT;   // 0,16,...,240
    const float invN  = 1.0f / (float)NN;

    const float* Wc = W + (size_t)c * NN * II * OO;

    // ---- Phase 0: Wr[n,i] = sum_o W[c,n,i,o] -> LDS ----------------------
    for (int idx = tid; idx < NN * II; idx += THREADS) {
        const float4* wp = (const float4*)(Wc + (size_t)idx * OO);
        float4 a = wp[0], b = wp[1], d = wp[2], e = wp[3];
        sWr[idx] = (a.x + a.y + a.z + a.w) + (b.x + b.y + b.z + b.w)
                 + (d.x + d.y + d.z + d.w) + (e.x + e.y + e.z + e.w);
    }
    __syncthreads();

    // ---- Phase 1: online softmax stats of l1[n] = r[b,n]/N ----------------
    {
        const int bq = tid >> 4, j = tid & 15;
        const float* xb = x + (size_t)(bbase + bq) * NN * II;
        float m = -3.4e38f, Z = 0.0f;
        for (int n = j; n < NN; n += 16) {
            const float4* xp = (const float4*)(xb + n * II);
            float4 xl = xp[0], xh = xp[1];
            const float* wr = sWr + n * II;
            float r = xl.x*wr[0] + xl.y*wr[1] + xl.z*wr[2] + xl.w*wr[3]
                    + xh.x*wr[4] + xh.y*wr[5] + xh.z*wr[6] + xh.w*wr[7];
            float l  = r * invN;
            float mn = fmaxf(m, l);
            Z = Z * __expf(m - mn) + __expf(l - mn);
            m = mn;
        }
        sRedM[bq][j] = m; sRedZ[bq][j] = Z;
    }
    __syncthreads();
    if (tid < 16) {
        float m = sRedM[tid][0], Z = sRedZ[tid][0];
        for (int k = 1; k < 16; ++k) {
            float mk = sRedM[tid][k], Zk = sRedZ[tid][k];
            float mn = fmaxf(m, mk);
            Z = Z * __expf(m - mn) + Zk * __expf(mk - mn);
            m = mn;
        }
        sM1[tid] = m; sZ1[tid] = Z;
    }
    __syncthreads();

    // ---- Phase 2: online softmax stats of b2[n] = l1 + c1[n]*r[n] ---------
    {
        const int bq = tid >> 4, j = tid & 15;
        const float m1 = sM1[bq], iZ1 = 1.0f / sZ1[bq];
        const float* xb = x + (size_t)(bbase + bq) * NN * II;
        float m = -3.4e38f, Z = 0.0f;
        for (int n = j; n < NN; n += 16) {
            const float4* xp = (const float4*)(xb + n * II);
            float4 xl = xp[0], xh = xp[1];
            const float* wr = sWr + n * II;
            float r = xl.x*wr[0] + xl.y*wr[1] + xl.z*wr[2] + xl.w*wr[3]
                    + xh.x*wr[4] + xh.y*wr[5] + xh.z*wr[6] + xh.w*wr[7];
            float l1 = r * invN;
            float c1 = __expf(l1 - m1) * iZ1;
            float b2 = l1 + c1 * r;
            float mn = fmaxf(m, b2);
            Z = Z * __expf(m - mn) + __expf(b2 - mn);
            m = mn;
        }
        sRedM[bq][j] = m; sRedZ[bq][j] = Z;
    }
    __syncthreads();
    if (tid < 16) {
        float m = sRedM[tid][0], Z = sRedZ[tid][0];
        for (int k = 1; k < 16; ++k) {
            float mk = sRedM[tid][k], Zk = sRedZ[tid][k];
            float mn = fmaxf(m, mk);
            Z = Z * __expf(m - mn) + Zk * __expf(mk - mn);
            m = mn;
        }
        sM2[tid] = m; sZ2[tid] = Z;
    }
    __syncthreads();

    // ---- Phase 3: weighted vote GEMM via V_WMMA_F32_16X16X4_F32 ----------
    // s[b,o] = sum_n c2[b,n] * sum_i x[b,n,i]*W[c,n,i,o]
    // Per n: M=16(b) x N=16(o) x K=8(i) -> two 16x16x4 f32 WMMAs.
    // A layout (16x4): VGPR j, lane-half g -> K = j + 2g, M = lane&15.
    // B layout (4x16): VGPR j, lane-half g -> K = j + 2g, o = lane&15.
    {
        const int lane = tid & 31;
        const int wave = tid >> 5;
        const int bq   = lane & 15;   // M row (b) for A; o column for B
        const int kg   = lane >> 4;   // lane-half -> K group
        const float m1 = sM1[bq], iZ1 = 1.0f / sZ1[bq];
        const float m2 = sM2[bq], iZ2 = 1.0f / sZ2[bq];
        const float* xb = x + (size_t)(bbase + bq) * NN * II;

        v8f acc = {0.f, 0.f, 0.f, 0.f, 0.f, 0.f, 0.f, 0.f};

        const int n0 = wave * NPW;
        for (int n = n0; n < n0 + NPW; ++n) {
            const float4* xp = (const float4*)(xb + n * II);
            float4 xl = xp[0], xh = xp[1];
            const float* wr = sWr + n * II;
            float r = xl.x*wr[0] + xl.y*wr[1] + xl.z*wr[2] + xl.w*wr[3]
                    + xh.x*wr[4] + xh.y*wr[5] + xh.z*wr[6] + xh.w*wr[7];
            float l1 = r * invN;
            float c1 = __expf(l1 - m1) * iZ1;
            float b2 = l1 + c1 * r;
            float cw = __expf(b2 - m2) * iZ2;   // final routing weight c2[b,n]

            const float* wn = Wc + (size_t)n * (II * OO);

            // WMMA 0: i = 0..3
            v2f A0 = { (kg ? xl.z : xl.x) * cw, (kg ? xl.w : xl.y) * cw };
            v2f B0 = { wn[(2*kg)     * OO + bq], wn[(2*kg + 1) * OO + bq] };
            acc = __builtin_amdgcn_wmma_f32_16x16x4_f32(
                      false, A0, false, B0, (short)0, acc, false, false);

            // WMMA 1: i = 4..7
            v2f A1 = { (kg ? xh.z : xh.x) * cw, (kg ? xh.w : xh.y) * cw };
            v2f B1 = { wn[(4 + 2*kg) * OO + bq], wn[(5 + 2*kg) * OO + bq] };
            acc = __builtin_amdgcn_wmma_f32_16x16x4_f32(
                      false, A1, false, B1, (short)0, acc, false, false);
        }
        for (int v = 0; v < 8; ++v) sAcc[wave][v][lane] = acc[v];
    }
    __syncthreads();

    // ---- Phase 4: cross-wave reduce, squash, write out[b,c,o] ------------
    {
        const int v  = tid >> 5;
        const int ln = tid & 31;
        float s = 0.0f;
        for (int w = 0; w < 8; ++w) s += sAcc[w][v][ln];
        const int M = v + 8 * (ln >> 4);   // C/D layout: VGPR v -> M=v | v+8
        const int o = ln & 15;
        sRedM[M][o] = s;                   // reuse as s[M][o]
        __syncthreads();
        float sq = 0.0f;
        for (int oo = 0; oo < 16; ++oo) { float t = sRedM[M][oo]; sq += t * t; }
        float factor = sq / ((1.0f + sq) * sqrtf(sq));   // squash scale
        out[(size_t)(bbase + M) * (CC * OO) + c * OO + o] = factor * s;
    }
}

extern "C" void kernel_launch(void* const* d_in, const int* in_sizes, int n_in,
                              void* d_out, int out_size, void* d_ws, size_t ws_size,
                              hipStream_t stream) {
    (void)in_sizes; (void)n_in; (void)d_ws; (void)ws_size; (void)out_size;
    const float* x = (const float*)d_in[0];   // [256,1152,8] f32
    const float* W = (const float*)d_in[1];   // [10,1152,8,16] f32
    float* out = (float*)d_out;               // [256,10,16] f32
    dim3 grid(CC * (BB / BT));                // 160 workgroups
    digitcaps_route_wmma<<<grid, THREADS, 0, stream>>>(x, W, out);
}